// Hierarical_CELoss2_82489141887106
// MI455X (gfx1250) — compile-verified
//
#include <hip/hip_runtime.h>
#include <cfloat>
#include <cstdint>

// ---- problem constants (from reference setup_inputs) ----
#define B_ROWS 32768
#define C_COLS 1000
#define ROWS_PER_BLK 8
#define MAIN_BLOCKS (B_ROWS / ROWS_PER_BLK)   // 4096
#define GAMMA_MAP 3.0f

typedef __attribute__((ext_vector_type(4))) unsigned int v4u;
typedef __attribute__((ext_vector_type(8))) int          v8i;
typedef __attribute__((ext_vector_type(4))) int          v4i;

// ---------------------------------------------------------------------------
// Kernel 0: zero the dmax accumulator (bits of a non-negative float)
// ---------------------------------------------------------------------------
__global__ void init_ws(unsigned int* ws_u) { ws_u[0] = 0u; }

// ---------------------------------------------------------------------------
// Kernel 1: global max of distance matrix (1e6 fp32, all >= 0)
// float bit pattern is monotone for non-negative floats -> atomicMax on uint.
// ---------------------------------------------------------------------------
__global__ __launch_bounds__(256) void dmax_kernel(const float* __restrict__ d,
                                                   unsigned int* __restrict__ ws_u) {
    const int NF4 = (C_COLS * C_COLS) / 4;     // 250000 float4
    const float4* d4 = (const float4*)d;
    float m = 0.0f;
    int tid    = blockIdx.x * blockDim.x + threadIdx.x;
    int stride = gridDim.x * blockDim.x;
    for (int i = tid; i < NF4; i += stride) {
        float4 v = d4[i];
        m = fmaxf(m, fmaxf(fmaxf(v.x, v.y), fmaxf(v.z, v.w)));
    }
    for (int off = 16; off > 0; off >>= 1)
        m = fmaxf(m, __shfl_xor(m, off, 32));
    __shared__ float sm[8];
    int lane = threadIdx.x & 31, wv = threadIdx.x >> 5;
    if (lane == 0) sm[wv] = m;
    __syncthreads();
    if (threadIdx.x == 0) {
        float bm = sm[0];
        #pragma unroll
        for (int i = 1; i < 8; ++i) bm = fmaxf(bm, sm[i]);
        atomicMax(ws_u, __float_as_uint(bm));
    }
}

// ---------------------------------------------------------------------------
// Kernel 2: per-row log-softmax-at-label + argmax + focal term.
// One block = 8 rows. One TDM async tensor load pulls the 8x1000 fp32 tile
// into LDS; wave w then reduces row w entirely with wave32 shuffles.
// ---------------------------------------------------------------------------
__global__ __launch_bounds__(256) void main_kernel(const float* __restrict__ y_pred,
                                                   const int*   __restrict__ y_true,
                                                   const float* __restrict__ dmat,
                                                   const unsigned int* __restrict__ ws_u,
                                                   float* __restrict__ part) {
    __shared__ float s_tile[ROWS_PER_BLK * C_COLS];   // 32000 B
    __shared__ float s_contrib[ROWS_PER_BLK];

    const int wv   = threadIdx.x >> 5;
    const int lane = threadIdx.x & 31;
    const int row0 = blockIdx.x * ROWS_PER_BLK;

    // ---- TDM: async DMA of the 8-row tile into LDS (issued by wave 0) ----
    if (threadIdx.x < 32) {
        unsigned long long gaddr =
            (unsigned long long)(uintptr_t)(const void*)(y_pred + (size_t)row0 * C_COLS);
        unsigned int lds_off = (unsigned int)(uintptr_t)(void*)s_tile; // LDS byte offset

        v4u g0;
        g0.x = 1u;                                            // count=1 (valid descriptor)
        g0.y = lds_off;                                       // lds_addr
        g0.z = (unsigned int)(gaddr & 0xFFFFFFFFull);         // global_addr[31:0]
        g0.w = (unsigned int)((gaddr >> 32) & 0x1FFFFFFull)   // global_addr[56:32]
             | (2u << 30);                                    // type=2 ("image")

        v8i g1;
        g1.s0 = (int)(2u << 16);                              // data_size=2 (4 bytes), mask=0
        g1.s1 = (int)((unsigned)C_COLS << 16);                // tensor_dim0[15:0] @ bits 63:48
        g1.s2 = (int)(((unsigned)B_ROWS & 0xFFFFu) << 16);    // tensor_dim1[15:0] @ bits 95:80
        g1.s3 = (int)((unsigned)C_COLS << 16);                // tile_dim0 @ bits 127:112
        g1.s4 = ROWS_PER_BLK;                                 // tile_dim1=8, tile_dim2=0
        g1.s5 = C_COLS;                                       // tensor_dim0_stride (elements)
        g1.s6 = 0;
        g1.s7 = 0;

        v4i g2 = {0, 0, 0, 0};                                // 2D tensor: groups 2/3 unused
        v4i g3 = {0, 0, 0, 0};
        v8i g4 = {0, 0, 0, 0, 0, 0, 0, 0};                    // extension group (unused)

        __builtin_amdgcn_tensor_load_to_lds(g0, g1, g2, g3, g4, 0);
        __builtin_amdgcn_s_wait_tensorcnt(0);
    }
    __syncthreads();

    // ---- wave w reduces row w from LDS, values cached in registers ----
    const float* my_row = s_tile + wv * C_COLS;
    float4 vals[8];
    float m  = -FLT_MAX;
    int   mi = 0;
    #pragma unroll
    for (int k = 0; k < 8; ++k) {
        int chunk = lane + 32 * k;                 // float4 chunk index, 250 valid
        if (chunk < C_COLS / 4) {
            vals[k] = *(const float4*)(my_row + chunk * 4);  // 16B-aligned ds_load_b128
            int c0 = chunk * 4;                    // ascending column order per lane
            if (vals[k].x > m) { m = vals[k].x; mi = c0;     }
            if (vals[k].y > m) { m = vals[k].y; mi = c0 + 1; }
            if (vals[k].z > m) { m = vals[k].z; mi = c0 + 2; }
            if (vals[k].w > m) { m = vals[k].w; mi = c0 + 3; }
        } else {
            vals[k] = make_float4(-FLT_MAX, -FLT_MAX, -FLT_MAX, -FLT_MAX);
        }
    }
    // wave32 butterfly max+argmax (smaller index wins ties -> first occurrence)
    for (int off = 16; off > 0; off >>= 1) {
        float om  = __shfl_xor(m,  off, 32);
        int   omi = __shfl_xor(mi, off, 32);
        if (om > m || (om == m && omi < mi)) { m = om; mi = omi; }
    }
    // sum of exp(x - max) from the register cache
    float s = 0.0f;
    #pragma unroll
    for (int k = 0; k < 8; ++k) {
        int chunk = lane + 32 * k;
        if (chunk < C_COLS / 4) {
            s += __expf(vals[k].x - m);
            s += __expf(vals[k].y - m);
            s += __expf(vals[k].z - m);
            s += __expf(vals[k].w - m);
        }
    }
    for (int off = 16; off > 0; off >>= 1) s += __shfl_xor(s, off, 32);

    if (lane == 0) {
        int   r     = row0 + wv;
        int   t     = y_true[r];
        float xt    = my_row[t];
        float logpt = xt - m - __logf(s);                    // log_softmax at label
        float dmaxv = __uint_as_float(ws_u[0]);              // d.max()
        float dv    = dmat[mi * C_COLS + t];                 // d[pred, label]
        float sim   = __expf(-GAMMA_MAP * dv / dmaxv);
        sim = fminf(fmaxf(sim, 0.0f), 1.0f);                 // clip to [0, 1]
        s_contrib[wv] = -(1.0f - sim) * logpt;               // focal term (gamma=1)
    }
    __syncthreads();
    if (threadIdx.x == 0) {
        float acc = 0.0f;
        #pragma unroll
        for (int i = 0; i < ROWS_PER_BLK; ++i) acc += s_contrib[i];
        part[blockIdx.x] = acc;                              // deterministic partials
    }
}

// ---------------------------------------------------------------------------
// Kernel 3: deterministic reduction of 4096 partials -> final scalar loss.
// loss = LAMBDA1 * mean(focal) + LAMBDA2 * ce ; LAMBDA1=0.5, LAMBDA2=0.
// ---------------------------------------------------------------------------
__global__ __launch_bounds__(256) void finalize_kernel(const float* __restrict__ part,
                                                       float* __restrict__ out) {
    float s = 0.0f;
    for (int i = threadIdx.x; i < MAIN_BLOCKS; i += 256) s += part[i];
    for (int off = 16; off > 0; off >>= 1) s += __shfl_xor(s, off, 32);
    __shared__ float sm[8];
    int lane = threadIdx.x & 31, wv = threadIdx.x >> 5;
    if (lane == 0) sm[wv] = s;
    __syncthreads();
    if (threadIdx.x == 0) {
        float acc = 0.0f;
        #pragma unroll
        for (int i = 0; i < 8; ++i) acc += sm[i];
        out[0] = 0.5f * acc / (float)B_ROWS;
    }
}

// ---------------------------------------------------------------------------
extern "C" void kernel_launch(void* const* d_in, const int* in_sizes, int n_in,
                              void* d_out, int out_size, void* d_ws, size_t ws_size,
                              hipStream_t stream) {
    const float* y_pred = (const float*)d_in[0];   // [32768, 1000] fp32
    const int*   y_true = (const int*)  d_in[1];   // [32768] int32
    const float* dmat   = (const float*)d_in[2];   // [1000, 1000] fp32

    unsigned int* ws_u = (unsigned int*)d_ws;                  // [0] = dmax bits
    float*        part = (float*)((char*)d_ws + 256);          // 4096 partials
    float*        out  = (float*)d_out;

    init_ws<<<1, 1, 0, stream>>>(ws_u);
    dmax_kernel<<<512, 256, 0, stream>>>(dmat, ws_u);
    main_kernel<<<MAIN_BLOCKS, 256, 0, stream>>>(y_pred, y_true, dmat, ws_u, part);
    finalize_kernel<<<1, 256, 0, stream>>>(part, out);
}